// baseline_9474697855596
// MI455X (gfx1250) — compile-verified
//
#include <hip/hip_runtime.h>
#include <math.h>

// ---- problem dims (from reference) ----
#define B_      32
#define NP1_    2001
#define NREG    2000      // NP1-1 regions
#define D_      512
#define R_      6
#define NROLES  190
#define NVERBS  504
#define VOCAB_  2001

typedef float v2f __attribute__((ext_vector_type(2)));
typedef float v8f __attribute__((ext_vector_type(8)));

// V_WMMA_F32_16X16X4_F32: full-precision f32 WMMA (keeps reference fp32 semantics).
static __device__ __forceinline__ v8f wmma4(v2f a, v2f b, v8f c) {
    return __builtin_amdgcn_wmma_f32_16x16x4_f32(false, a, false, b, (short)0, c,
                                                 false, false);
}

// ---------------------------------------------------------------------------
// Kernel 1: role_emb[r,d] = roles[r,:]·Wr[d,:] + br[d]   (6x512, K=190)
//           s_role[r]    = role_emb[r,:]·Wa[0,0:512]
// ---------------------------------------------------------------------------
__global__ void role_kernel(const float* __restrict__ roles,
                            const float* __restrict__ Wr,
                            const float* __restrict__ br,
                            const float* __restrict__ Wa,
                            float* __restrict__ role_emb,
                            float* __restrict__ s_role) {
    __shared__ float red[256];
    const int tid = threadIdx.x;
    for (int idx = tid; idx < R_ * D_; idx += 256) {
        int r = idx / D_, d = idx % D_;
        float acc = br[d];
        for (int k = 0; k < NROLES; ++k)
            acc += roles[r * NROLES + k] * Wr[d * NROLES + k];
        role_emb[idx] = acc;
    }
    __syncthreads();
    for (int r = 0; r < R_; ++r) {
        float p = 0.f;
        for (int d = tid; d < D_; d += 256) p += role_emb[r * D_ + d] * Wa[d];
        red[tid] = p; __syncthreads();
        for (int s = 128; s > 0; s >>= 1) {
            if (tid < s) red[tid] += red[tid + s];
            __syncthreads();
        }
        if (tid == 0) s_role[r] = red[0];
        __syncthreads();
    }
}

// ---------------------------------------------------------------------------
// Kernel 2: verb_predict = relu(vs[:,0,:]) @ Wv.T + bv   (M=32, N=504, K=512)
// Out-of-range columns read a CLAMPED row of Wv (unconditional b64 loads, no
// exec divergence); their garbage C columns are simply never stored.
// ---------------------------------------------------------------------------
__global__ void verb_kernel(const float* __restrict__ vs,
                            const float* __restrict__ Wv,
                            const float* __restrict__ bv,
                            float* __restrict__ out) {
    const int lane = threadIdx.x & 31, wave = threadIdx.x >> 5;
    const int hi = (lane >> 4) & 1, l15 = lane & 15;
    const int m = blockIdx.x * 16 + l15;                 // batch row (<32)
    const int n = blockIdx.y * 64 + wave * 16 + l15;     // verb col
    const int nc = n < NVERBS ? n : NVERBS - 1;          // clamped (always valid)
    const float* arow = vs + (size_t)m * NP1_ * D_;      // vs[m,0,:]
    const float* brow = Wv + (size_t)nc * D_;
    v8f c0 = {}, c1 = {};
    for (int kb = 0; kb < D_; kb += 8) {
        const int k0 = kb + 2 * hi;
        const int k1 = kb + 4 + 2 * hi;
        v2f a0, a1, b0, b1;
        float x0 = arow[k0], x1 = arow[k0 + 1];
        float y0 = arow[k1], y1 = arow[k1 + 1];
        a0.x = x0 > 0.f ? x0 : 0.f;  a0.y = x1 > 0.f ? x1 : 0.f;
        a1.x = y0 > 0.f ? y0 : 0.f;  a1.y = y1 > 0.f ? y1 : 0.f;
        b0.x = brow[k0]; b0.y = brow[k0 + 1];
        b1.x = brow[k1]; b1.y = brow[k1 + 1];
        c0 = wmma4(a0, b0, c0);
        c1 = wmma4(a1, b1, c1);
    }
    if (n < NVERBS) {
        float bias = bv[n];
#pragma unroll
        for (int v = 0; v < 8; ++v) {
            int mm = blockIdx.x * 16 + v + 8 * hi;       // <= 31
            out[(size_t)mm * NVERBS + n] = c0[v] + c1[v] + bias;
        }
    }
}

// ---------------------------------------------------------------------------
// Kernel 3: s_vert[b,n] = regions[b,n,:]·Wa[0,512:1024]
// One wave per region row; the roofline-critical HBM sweep -> float4 loads.
// ---------------------------------------------------------------------------
__global__ void svert_kernel(const float* __restrict__ vs,
                             const float* __restrict__ Wa,
                             float* __restrict__ s_vert) {
    const int wid = (blockIdx.x * blockDim.x + threadIdx.x) >> 5;
    const int lane = threadIdx.x & 31;
    if (wid >= B_ * NREG) return;
    const int b = wid / NREG, n = wid % NREG;
    const float4* row = (const float4*)(vs + ((size_t)b * NP1_ + 1 + n) * D_);
    const float4* wv  = (const float4*)(Wa + D_);        // Wa_vert
    float p = 0.f;
#pragma unroll
    for (int i = 0; i < D_ / 128; ++i) {                 // 4 x b128 per lane
        float4 r4 = row[lane + i * 32];
        float4 w4 = wv[lane + i * 32];
        p += r4.x * w4.x + r4.y * w4.y + r4.z * w4.z + r4.w * w4.w;
    }
    for (int o = 16; o > 0; o >>= 1) p += __shfl_down(p, o, 32);
    if (lane == 0) s_vert[wid] = p;
}

// ---------------------------------------------------------------------------
// Kernel 4: att[b,r,n] = log_softmax_n( tanh(s_role[r] + s_vert[b,n] + ba) )
// ---------------------------------------------------------------------------
__global__ void att_kernel(const float* __restrict__ s_vert,
                           const float* __restrict__ s_role,
                           const float* __restrict__ ba,
                           float* __restrict__ att) {
    __shared__ float sv[NREG];
    __shared__ float sc[NREG];
    __shared__ float red[256];
    const int b = blockIdx.x, tid = threadIdx.x;
    for (int n = tid; n < NREG; n += 256) sv[n] = s_vert[b * NREG + n];
    const float bav = ba[0];
    __syncthreads();
    for (int r = 0; r < R_; ++r) {
        const float sr = s_role[r];
        float lmax = -1e30f;
        for (int n = tid; n < NREG; n += 256) {
            float t = tanhf(sr + sv[n] + bav);
            sc[n] = t;
            lmax = fmaxf(lmax, t);
        }
        red[tid] = lmax; __syncthreads();
        for (int s = 128; s > 0; s >>= 1) {
            if (tid < s) red[tid] = fmaxf(red[tid], red[tid + s]);
            __syncthreads();
        }
        const float mx = red[0]; __syncthreads();
        float lsum = 0.f;
        for (int n = tid; n < NREG; n += 256) lsum += expf(sc[n] - mx);
        red[tid] = lsum; __syncthreads();
        for (int s = 128; s > 0; s >>= 1) {
            if (tid < s) red[tid] += red[tid + s];
            __syncthreads();
        }
        const float lse = mx + logf(red[0]); __syncthreads();
        for (int n = tid; n < NREG; n += 256)
            att[((size_t)b * R_ + r) * NREG + n] = sc[n] - lse;
    }
}

// ---------------------------------------------------------------------------
// Kernel 5: label_embed[b] = att[b] (6x2000) @ regions[b] (2000x512)
// A padded to 16 rows by CLAMPING the row index (row m of A only touches row m
// of C; rows >= 6 are computed-but-never-stored). Second sweep hits L2
// (131 MB vert_states < 192 MB global L2).
// ---------------------------------------------------------------------------
__global__ void lembed_kernel(const float* __restrict__ vs,
                              const float* __restrict__ att,
                              float* __restrict__ le) {
    const int lane = threadIdx.x & 31, wave = threadIdx.x >> 5;
    const int hi = (lane >> 4) & 1, l15 = lane & 15;
    const int b = blockIdx.x;
    const int n = blockIdx.y * 128 + wave * 16 + l15;    // < 512 always
    const int arow = l15 < R_ ? l15 : R_ - 1;            // clamped A row
    const float* attr = att + ((size_t)b * R_ + arow) * NREG;
    const float* regb = vs + ((size_t)b * NP1_ + 1) * D_;
    v8f c0 = {}, c1 = {};
    for (int kb = 0; kb < NREG; kb += 8) {
        const int k0 = kb + 2 * hi;
        const int k1 = kb + 4 + 2 * hi;
        v2f a0, a1, b0, b1;
        a0.x = attr[k0]; a0.y = attr[k0 + 1];
        a1.x = attr[k1]; a1.y = attr[k1 + 1];
        b0.x = regb[(size_t)k0 * D_ + n];
        b0.y = regb[(size_t)(k0 + 1) * D_ + n];
        b1.x = regb[(size_t)k1 * D_ + n];
        b1.y = regb[(size_t)(k1 + 1) * D_ + n];
        c0 = wmma4(a0, b0, c0);
        c1 = wmma4(a1, b1, c1);
    }
#pragma unroll
    for (int v = 0; v < 8; ++v) {
        int m = v + 8 * hi;
        if (m < R_) le[((size_t)b * R_ + m) * D_ + n] = c0[v] + c1[v];
    }
}

// ---------------------------------------------------------------------------
// Kernel 6: role_label = relu(label_embed) (192x512) @ Wl.T (512x2001) + bl
// Clamped B row for n >= 2001; wave-uniform skip of fully-OOB tiles.
// ---------------------------------------------------------------------------
__global__ void rlabel_kernel(const float* __restrict__ le,
                              const float* __restrict__ Wl,
                              const float* __restrict__ bl,
                              float* __restrict__ out) {
    const int lane = threadIdx.x & 31, wave = threadIdx.x >> 5;
    const int hi = (lane >> 4) & 1, l15 = lane & 15;
    const int ct = blockIdx.y * 8 + wave;                // column tile 0..127
    const int n0 = ct * 16;
    if (n0 >= VOCAB_) return;                            // wave-uniform skip
    const int m = blockIdx.x * 16 + l15;                 // < 192 always
    const int n = n0 + l15;
    const int nc = n < VOCAB_ ? n : VOCAB_ - 1;          // clamped (always valid)
    const float* arow = le + (size_t)m * D_;
    const float* brow = Wl + (size_t)nc * D_;
    v8f c0 = {}, c1 = {};
    for (int kb = 0; kb < D_; kb += 8) {
        const int k0 = kb + 2 * hi;
        const int k1 = kb + 4 + 2 * hi;
        v2f a0, a1, b0, b1;
        float x0 = arow[k0], x1 = arow[k0 + 1];
        float y0 = arow[k1], y1 = arow[k1 + 1];
        a0.x = x0 > 0.f ? x0 : 0.f;  a0.y = x1 > 0.f ? x1 : 0.f;  // relu fused
        a1.x = y0 > 0.f ? y0 : 0.f;  a1.y = y1 > 0.f ? y1 : 0.f;
        b0.x = brow[k0]; b0.y = brow[k0 + 1];
        b1.x = brow[k1]; b1.y = brow[k1 + 1];
        c0 = wmma4(a0, b0, c0);
        c1 = wmma4(a1, b1, c1);
    }
    if (n < VOCAB_) {
        float bias = bl[n];
#pragma unroll
        for (int v = 0; v < 8; ++v) {
            int mm = blockIdx.x * 16 + v + 8 * hi;       // < 192
            out[(size_t)mm * VOCAB_ + n] = c0[v] + c1[v] + bias;
        }
    }
}

// ---------------------------------------------------------------------------
extern "C" void kernel_launch(void* const* d_in, const int* in_sizes, int n_in,
                              void* d_out, int out_size, void* d_ws, size_t ws_size,
                              hipStream_t stream) {
    const float* vs    = (const float*)d_in[0];  // (32,2001,512)
    const float* roles = (const float*)d_in[1];  // (6,190)
    const float* Wr    = (const float*)d_in[2];  // (512,190)
    const float* br    = (const float*)d_in[3];  // (512)
    const float* Wa    = (const float*)d_in[4];  // (1,1024)
    const float* ba    = (const float*)d_in[5];  // (1)
    const float* Wv    = (const float*)d_in[6];  // (504,512)
    const float* bv    = (const float*)d_in[7];  // (504)
    const float* Wl    = (const float*)d_in[8];  // (2001,512)
    const float* bl    = (const float*)d_in[9];  // (2001)

    float* out_verb = (float*)d_out;             // 32*504
    float* out_rl   = out_verb + B_ * NVERBS;    // 32*6*2001

    // workspace layout (floats): ~2.2 MB total
    float* ws       = (float*)d_ws;
    float* role_emb = ws;                        // 3072
    float* s_role   = ws + 3072;                 // 16 (padded)
    float* s_vert   = ws + 3072 + 16;            // 64000
    float* att      = s_vert + 64000;            // 384000
    float* le       = att + 384000;              // 98304

    role_kernel  <<<1,             256, 0, stream>>>(roles, Wr, br, Wa, role_emb, s_role);
    verb_kernel  <<<dim3(2, 8),    128, 0, stream>>>(vs, Wv, bv, out_verb);
    svert_kernel <<<8000,          256, 0, stream>>>(vs, Wa, s_vert);
    att_kernel   <<<B_,            256, 0, stream>>>(s_vert, s_role, ba, att);
    lembed_kernel<<<dim3(B_, 4),   256, 0, stream>>>(vs, att, le);
    rlabel_kernel<<<dim3(12, 16),  256, 0, stream>>>(le, Wl, bl, out_rl);
}